// Attention_59493886984284
// MI455X (gfx1250) — compile-verified
//
#include <hip/hip_runtime.h>
#include <math.h>

typedef __attribute__((ext_vector_type(2))) float v2f;
typedef __attribute__((ext_vector_type(8))) float v8f;

#define QN 2048
#define KN 2048
#define CIN 256
#define HD 256
#define NH 8
#define CH 32
#define KSPLIT 4

__device__ __forceinline__ v8f wmma_f32x4(v2f a, v2f b, v8f c) {
    return __builtin_amdgcn_wmma_f32_16x16x4_f32(
        /*neg_a=*/false, a, /*neg_b=*/false, b,
        /*c_mod=*/(short)0, c, /*reuse_a=*/false, /*reuse_b=*/false);
}

// D(16x16) = A(16xCK, row-major lda) * B(CKx16, row-major ldb), f32 WMMA.
__device__ __forceinline__ v8f gemm_tile_f32(const float* __restrict__ A, int lda,
                                             const float* __restrict__ B, int ldb,
                                             int CK, int lane) {
    const int m = lane & 15;
    const int hi = lane >> 4;   // lane half selects K pair within chunk of 4
    const int n = m;
    v8f acc = {};
    for (int k0 = 0; k0 < CK; k0 += 4) {
        v2f a, b;
        a.x = A[m * lda + k0 + 2 * hi + 0];
        a.y = A[m * lda + k0 + 2 * hi + 1];
        b.x = B[(k0 + 2 * hi + 0) * ldb + n];
        b.y = B[(k0 + 2 * hi + 1) * ldb + n];
        acc = wmma_f32x4(a, b, acc);
    }
    return acc;
}

// ---------------- Kernel 1: fused projections (q*scale, k, v, sigmoid gate) --
__global__ void proj_kernel(const float* __restrict__ qx, const float* __restrict__ kvx,
                            const float* __restrict__ Wq, const float* __restrict__ Wk,
                            const float* __restrict__ Wv, const float* __restrict__ Wg,
                            const float* __restrict__ bg,
                            float* __restrict__ q_ws, float* __restrict__ k_ws,
                            float* __restrict__ v_ws, float* __restrict__ g_ws) {
    const int lane = threadIdx.x & 31;
    const int tile = blockIdx.x;        // 0..15 column tiles of 16
    const int row0 = blockIdx.y * 16;   // 0..127 row tiles
    const int which = blockIdx.z;       // 0=q 1=k 2=v 3=gate

    const float* A = (which == 1 || which == 2) ? kvx : qx;
    const float* W = (which == 0) ? Wq : (which == 1) ? Wk : (which == 2) ? Wv : Wg;
    float* O = (which == 0) ? q_ws : (which == 1) ? k_ws : (which == 2) ? v_ws : g_ws;

    v8f acc = gemm_tile_f32(A + (size_t)row0 * CIN, CIN, W + tile * 16, HD, CIN, lane);

    const int m = lane & 15, hi = lane >> 4, n = m;
    const int col = tile * 16 + n;
    const float bgc = (which == 3) ? bg[col] : 0.f;
#pragma unroll
    for (int r = 0; r < 8; ++r) {
        float val = acc[r];
        if (which == 0) val *= 0.17677669529663687f;  // 1/sqrt(32)
        if (which == 3) val = __builtin_amdgcn_rcpf(1.f + __expf(-(val + bgc)));
        O[(size_t)(row0 + r + 8 * hi) * HD + col] = val;
    }
}

// ---------------- Kernel 2: flash attention, split-K x4 + LDS merge ----------
// One block = one (head, 16-row q tile); 4 waves each own a 512-col K slice.
__global__ void __launch_bounds__(128)
flash_kernel(const float* __restrict__ q_ws, const float* __restrict__ k_ws,
             const float* __restrict__ v_ws, const float* __restrict__ g_ws,
             const float* __restrict__ bias_mask, const float* __restrict__ bias_pair,
             float* __restrict__ o_ws) {
    __shared__ float ldsP[KSPLIT][16 * 18];  // padded P relayout slabs (per wave)
    __shared__ float Ms[KSPLIT][16];         // per-slice row max
    __shared__ float Ls[KSPLIT][16];         // per-slice softmax denominator
    __shared__ float O0s[KSPLIT][16][16];    // per-slice O partial, c_hid [0,16)
    __shared__ float O1s[KSPLIT][16][16];    // per-slice O partial, c_hid [16,32)

    const int lane = threadIdx.x & 31;
    const int wv = threadIdx.x >> 5;         // K-slice index 0..3
    const int h = blockIdx.x >> 7;           // head 0..7
    const int q0 = (blockIdx.x & 127) * 16;  // q tile base row
    const int m = lane & 15, hi = lane >> 4, n = m;
    float* P = ldsP[wv];

    // Q A-fragments for all 8 K-chunks (loop invariant, q pre-scaled).
    v2f aq[8];
    const float* Qb = q_ws + (size_t)q0 * HD + h * CH;
#pragma unroll
    for (int kk = 0; kk < 8; ++kk) {
        aq[kk].x = Qb[m * HD + kk * 4 + 2 * hi + 0];
        aq[kk].y = Qb[m * HD + kk * 4 + 2 * hi + 1];
    }

    v8f o0 = {}, o1 = {}, o2 = {};  // o2: rowsum-of-P accumulator (ones trick)
    float mrow[8];
#pragma unroll
    for (int r = 0; r < 8; ++r) mrow[r] = -1e30f;

    v2f bones; bones.x = 1.f; bones.y = 1.f;  // all-ones B fragment
    const float* bp_base = bias_pair + (size_t)h * QN * KN + (size_t)(q0 + 8 * hi) * KN + n;
    const int kb_lo = wv * (KN / KSPLIT);
    const int kb_hi = kb_lo + (KN / KSPLIT);

    for (int kb = kb_lo; kb < kb_hi; kb += 16) {
        // ---- S tile = Q * K^T (16x16), inner 32 in chunks of 4
        v8f s = {};
#pragma unroll
        for (int kk = 0; kk < 8; ++kk) {
            const float* Kb = k_ws + (size_t)(kb + n) * HD + h * CH + kk * 4 + 2 * hi;
            v2f b; b.x = Kb[0]; b.y = Kb[1];
            s = wmma_f32x4(aq[kk], b, s);
        }
        // ---- additive biases (bias_pair = the HBM-bound stream: non-temporal)
        const float bm = bias_mask[kb + n];
        const float* bp = bp_base + kb;
#pragma unroll
        for (int r = 0; r < 8; ++r)
            s[r] += bm + __builtin_nontemporal_load(bp + (size_t)r * KN);
        // ---- online softmax: row max via 16-lane xor-reduce; sums via o2 WMMA
#pragma unroll
        for (int r = 0; r < 8; ++r) {
            float mt = s[r];
            mt = fmaxf(mt, __shfl_xor(mt, 1, 32));
            mt = fmaxf(mt, __shfl_xor(mt, 2, 32));
            mt = fmaxf(mt, __shfl_xor(mt, 4, 32));
            mt = fmaxf(mt, __shfl_xor(mt, 8, 32));
            const float mnew = fmaxf(mrow[r], mt);
            const float corr = __expf(mrow[r] - mnew);
            mrow[r] = mnew;
            s[r] = __expf(s[r] - mnew);
            o0[r] *= corr;
            o1[r] *= corr;
            o2[r] *= corr;
        }
        // ---- relayout P (C/D layout -> A layout) via private LDS slab
#pragma unroll
        for (int r = 0; r < 8; ++r) P[(r + 8 * hi) * 18 + n] = s[r];
        // ---- O += P*V and o2 += P*ones (softmax denominator on the matrix pipe)
#pragma unroll
        for (int kk = 0; kk < 4; ++kk) {
            v2f ap;
            ap.x = P[m * 18 + kk * 4 + 2 * hi + 0];
            ap.y = P[m * 18 + kk * 4 + 2 * hi + 1];
            const float* Vb = v_ws + (size_t)(kb + kk * 4 + 2 * hi) * HD + h * CH;
            v2f b0, b1;
            b0.x = Vb[n];        b0.y = Vb[HD + n];
            b1.x = Vb[16 + n];   b1.y = Vb[HD + 16 + n];
            o0 = wmma_f32x4(ap, b0, o0);
            o1 = wmma_f32x4(ap, b1, o1);
            o2 = wmma_f32x4(ap, bones, o2);
        }
    }

    // ---- publish per-slice partials (m replicated in half; o2 = rowsum in every col)
    if (n == 0) {
#pragma unroll
        for (int r = 0; r < 8; ++r) {
            Ms[wv][r + 8 * hi] = mrow[r];
            Ls[wv][r + 8 * hi] = o2[r];
        }
    }
#pragma unroll
    for (int r = 0; r < 8; ++r) {
        O0s[wv][r + 8 * hi][n] = o0[r];
        O1s[wv][r + 8 * hi][n] = o1[r];
    }
    __syncthreads();

    // ---- wave 0 merges slices, applies 1/l and sigmoid gate, stores [Q,HD]
    if (wv == 0) {
        const float* gb = g_ws + (size_t)q0 * HD + h * CH;
        float* ob = o_ws + (size_t)q0 * HD + h * CH;
#pragma unroll
        for (int r = 0; r < 8; ++r) {
            const int row = r + 8 * hi;
            const float M = fmaxf(fmaxf(Ms[0][row], Ms[1][row]),
                                  fmaxf(Ms[2][row], Ms[3][row]));
            float l = 0.f, a0 = 0.f, a1 = 0.f;
#pragma unroll
            for (int w = 0; w < KSPLIT; ++w) {
                const float f = __expf(Ms[w][row] - M);
                l  += Ls[w][row] * f;
                a0 += O0s[w][row][n] * f;
                a1 += O1s[w][row][n] * f;
            }
            const float inv = __builtin_amdgcn_rcpf(l);
            ob[(size_t)row * HD + n]      = a0 * inv * gb[(size_t)row * HD + n];
            ob[(size_t)row * HD + 16 + n] = a1 * inv * gb[(size_t)row * HD + 16 + n];
        }
    }
}

// ---------------- Kernel 3: output projection out = o_ws @ Wo + bo ------------
__global__ void out_kernel(const float* __restrict__ o_ws, const float* __restrict__ Wo,
                           const float* __restrict__ bo, float* __restrict__ out) {
    const int lane = threadIdx.x & 31;
    const int tile = blockIdx.x;       // 16 col tiles
    const int row0 = blockIdx.y * 16;  // 128 row tiles
    v8f acc = gemm_tile_f32(o_ws + (size_t)row0 * HD, HD, Wo + tile * 16, CIN, HD, lane);
    const int m = lane & 15, hi = lane >> 4, n = m;
    const float b = bo[tile * 16 + n];
#pragma unroll
    for (int r = 0; r < 8; ++r)
        out[(size_t)(row0 + r + 8 * hi) * CIN + tile * 16 + n] = acc[r] + b;
}

extern "C" void kernel_launch(void* const* d_in, const int* in_sizes, int n_in,
                              void* d_out, int out_size, void* d_ws, size_t ws_size,
                              hipStream_t stream) {
    const float* q_x       = (const float*)d_in[0];
    const float* kv_x      = (const float*)d_in[1];
    const float* bias_mask = (const float*)d_in[2];
    const float* bias_pair = (const float*)d_in[3];
    const float* Wq        = (const float*)d_in[4];
    const float* Wk        = (const float*)d_in[5];
    const float* Wv        = (const float*)d_in[6];
    const float* Wo        = (const float*)d_in[7];
    const float* bo        = (const float*)d_in[8];
    const float* Wg        = (const float*)d_in[9];
    const float* bg        = (const float*)d_in[10];
    float* out = (float*)d_out;

    const size_t mat = (size_t)QN * HD; // 2048*256 floats per workspace matrix
    float* q_ws = (float*)d_ws;
    float* k_ws = q_ws + mat;
    float* v_ws = k_ws + mat;
    float* g_ws = v_ws + mat;
    float* o_ws = g_ws + mat;

    // Kernel 1: all four projections. One wave per 16x16 tile.
    proj_kernel<<<dim3(HD / 16, QN / 16, 4), 32, 0, stream>>>(
        q_x, kv_x, Wq, Wk, Wv, Wg, bg, q_ws, k_ws, v_ws, g_ws);

    // Kernel 2: flash attention; one block per (head, q-tile), 4-way split-K.
    flash_kernel<<<dim3(NH * (QN / 16)), 128, 0, stream>>>(
        q_ws, k_ws, v_ws, g_ws, bias_mask, bias_pair, o_ws);

    // Kernel 3: output projection.
    out_kernel<<<dim3(CIN / 16, QN / 16), 32, 0, stream>>>(o_ws, Wo, bo, out);
}